// TransPWCLite_67499706024522
// MI455X (gfx1250) — compile-verified
//
#include <hip/hip_runtime.h>

// -------------------------------------------------------------------------
// PWC-Net cost volume on gfx1250 via V_WMMA_F32_16X16X4_F32.
//
// cost[b, s=(di+4)*9+(dj+4), h, w] = (1/128) * sum_c f1[b,c,h,w]*f2[b,c,h+di,w+dj]
// B=4, C=128, H=256, W=448, max_disp=4 (81 shifts), zero padding.
//
// Banded-GEMM mapping: A tile (M=16) packs TWO h rows x 8 w pixels:
//   M 0..7  -> pixels (h,   w0..w0+7)
//   M 8..15 -> pixels (h+1, w0..w0+7)
// B tile (N=16) = feat2 row r, columns w0-4 .. w0+11, K = C = 128.
// D[m,n] then contains, on its band:
//   m<8 : cost(pixel (h,  w0+m),  di=r-h,   dj=n-m-4)      all 9 dj in-window
//   m>=8: cost(pixel (h+1,w0+m-8),di=r-h-1, dj=n-(m-8)-4)  all 9 dj in-window
// Sweeping r over h-4..h+5 covers all 81 shifts: 56% WMMA slot utilization
// (vs 28% for the naive 16-pixel strip), full fp32 precision.
//
// Workgroup: 128 threads = 4 waves, each wave owns one h-pair (h0+2*wid),
// all share one 8-wide w strip. feat2 rows h0-4..h0+11 (16 rows x 16 cols x
// 128 ch = 128 KB) are staged once in LDS in WMMA-B operand order, so each
// lane's per-step B operand is a single aligned ds_load_b64. A is held in 64
// VGPRs per lane for the whole kernel. Results go through an LDS out-stage
// (bank-conflict-free: dword stride 81, odd mod 64) and are stored coalesced.
// -------------------------------------------------------------------------

typedef float v2f __attribute__((ext_vector_type(2)));
typedef float v8f __attribute__((ext_vector_type(8)));
typedef float f32x4 __attribute__((ext_vector_type(4)));

#define C_CH 128
#define HDIM 256
#define WDIM 448
#define NSHIFT 81

// LDS layout:
//  Bst: 16 rows x 2048 floats. Row layout (operand order for wmma f32 16x16x4):
//       element (k, col) at dword ((k>>1)*16 + col)*2 + (k&1)
//       -> lane reading (k0, k0+1) at column col gets one aligned float2.
//  Ost: [hp_local(8)][wp(8)][s(81)] floats.
#define B_ROW_DW 2048
#define OST_OFF  (16 * B_ROW_DW)
#define OST_N    (8 * 8 * NSHIFT)

__global__ __launch_bounds__(128)
void pwc_cost_volume_wmma(const float* __restrict__ f1,
                          const float* __restrict__ f2,
                          float* __restrict__ out)
{
    extern __shared__ float smem[];
    float* Bst = smem;
    float* Ost = smem + OST_OFF;

    const int tid  = threadIdx.x;
    const int lane = tid & 31;
    const int wid  = tid >> 5;
    const int w0   = blockIdx.x * 8;
    const int h0   = blockIdx.y * 8;
    const int b    = blockIdx.z;

    const long plane = (long)HDIM * WDIM;

    // ---- zero the output stage (covers zero-padded / skipped shifts) ----
    for (int e = tid; e < OST_N; e += 128) Ost[e] = 0.0f;

    // ---- stage feat2: 16 rows (h0-4..h0+11) x 16 cols (w0-4..w0+11) x 128 ch
    for (int c = tid; c < 8192; c += 128) {      // float4 chunks
        const int col4 = (c & 3) << 2;           // 0,4,8,12
        const int k    = (c >> 2) & 127;
        const int row  = c >> 9;                 // 0..15
        const int rabs = h0 - 4 + row;
        f32x4 v = {0.f, 0.f, 0.f, 0.f};
        if ((unsigned)rabs < (unsigned)HDIM) {
            const int cbase = w0 - 4 + col4;
            const float* src = f2 + ((long)(b * C_CH + k) * HDIM + rabs) * WDIM + cbase;
            if (cbase >= 0 && cbase + 3 < WDIM) {
                v = *(const f32x4*)src;          // 16B aligned: w0-4 is mult of 4
            } else {
                #pragma unroll
                for (int j = 0; j < 4; ++j) {
                    const int cc = cbase + j;
                    if ((unsigned)cc < (unsigned)WDIM) v[j] = src[j];
                }
            }
        }
        float* dst = Bst + row * B_ROW_DW + ((((k >> 1) * 16 + col4) << 1) + (k & 1));
        dst[0] = v[0]; dst[2] = v[1]; dst[4] = v[2]; dst[6] = v[3];
    }

    // ---- load this wave's A tile into 64 VGPRs per lane ----
    // A layout (fp32 16x4): lane L holds M = L&15, K-pair offset 2*(L>>4).
    const int m    = lane & 15;
    const int tt   = lane >> 4;                  // 0/1
    const int habs = h0 + wid * 2;               // wave's first h row
    const float* asrc = f1 + (long)b * C_CH * plane
                           + (long)(habs + (m >> 3)) * WDIM + (w0 + (m & 7));
    float a[64];
    #pragma unroll
    for (int s = 0; s < 32; ++s) {
        const int k0 = 4 * s + 2 * tt;
        a[2 * s]     = asrc[(long)k0 * plane];
        a[2 * s + 1] = asrc[(long)(k0 + 1) * plane];
    }

    __syncthreads();

    // ---- main loop: one WMMA K-chain per staged feat2 row this wave uses ----
    for (int row = 0; row < 16; ++row) {
        const int rabs = h0 - 4 + row;
        const int dr = rabs - habs;              // di for the m<8 half
        if (dr < -4 || dr > 5) continue;         // row not used by this wave
        if ((unsigned)rabs >= (unsigned)HDIM) continue;  // zero padding -> stays 0

        const float* brow = Bst + row * B_ROW_DW;
        v8f acc = {};
        #pragma unroll
        for (int s = 0; s < 32; ++s) {
            // B operand: lane L holds N = L&15, K-pair offset 2*(L>>4).
            const v2f bop = *(const v2f*)(brow + ((((s * 2 + tt) * 16) + m) << 1));
            const v2f aop = { a[2 * s], a[2 * s + 1] };
            acc = __builtin_amdgcn_wmma_f32_16x16x4_f32(
                false, aop, false, bop, (short)0, acc, false, false);
        }

        // ---- scatter band entries to the LDS out-stage ----
        // acc VGPR g: M = g + 8*(lane>>4), N = lane&15
        const int nIdx = lane & 15;
        const int hpo  = lane >> 4;              // which h row of the pair
        const int di   = dr - hpo;
        if (di >= -4 && di <= 4) {
            #pragma unroll
            for (int g = 0; g < 8; ++g) {
                const int dj = nIdx - g - 4;
                if (dj >= -4 && dj <= 4) {
                    const int sidx = (di + 4) * 9 + (dj + 4);
                    Ost[((wid * 2 + hpo) * 8 + g) * NSHIFT + sidx] = acc[g] * (1.0f / 128.0f);
                }
            }
        }
    }

    __syncthreads();

    // ---- coalesced writeback: 32B contiguous segments per (s, hp) ----
    for (int e = tid; e < OST_N; e += 128) {
        const int wp = e & 7;
        const int hp = (e >> 3) & 7;
        const int s  = e >> 6;
        out[(((long)b * NSHIFT + s) * HDIM + (h0 + hp)) * WDIM + (w0 + wp)] =
            Ost[(hp * 8 + wp) * NSHIFT + s];
    }
}

extern "C" void kernel_launch(void* const* d_in, const int* in_sizes, int n_in,
                              void* d_out, int out_size, void* d_ws, size_t ws_size,
                              hipStream_t stream) {
    (void)in_sizes; (void)n_in; (void)d_ws; (void)ws_size; (void)out_size;
    const float* f1 = (const float*)d_in[0];
    const float* f2 = (const float*)d_in[1];
    // d_in[2] = max_disp scalar (fixed at 4; specialized at compile time)
    float* out = (float*)d_out;

    const size_t shmem = (size_t)(OST_OFF + OST_N) * sizeof(float);  // ~148 KB
    // Opt in to >default dynamic LDS (no-op on stacks that allow it natively).
    (void)hipFuncSetAttribute((const void*)pwc_cost_volume_wmma,
                              hipFuncAttributeMaxDynamicSharedMemorySize,
                              (int)shmem);

    dim3 grid(WDIM / 8, HDIM / 8, 4);   // 56 x 32 x 4 workgroups
    dim3 block(128);                    // 4 waves: one h-pair each
    pwc_cost_volume_wmma<<<grid, block, shmem, stream>>>(f1, f2, out);
}